// additive_attention_51926154609340
// MI455X (gfx1250) — compile-verified
//
#include <hip/hip_runtime.h>
#include <math.h>

typedef __attribute__((ext_vector_type(2))) float v2f;
typedef __attribute__((ext_vector_type(8))) float v8f;

#define NB 4
#define LQ 512
#define LK 512
#define QF 256
#define HD 128
#define VD 256

#define QROWS 8
#define KCH 256
#define KPAD 257   // 256 keys + 1 pad float -> conflict-free transposed stage

// ---------------------------------------------------------------------------
// Kernel 1: fused Q and K projections, f32 WMMA 16x16x4.
// One wave computes a 16-row x 128-col (full H) output tile.
// Waves 0..127 -> queries*Wq^T, waves 128..255 -> keys*Wk^T.
// ---------------------------------------------------------------------------
__global__ __launch_bounds__(256) void proj_kernel(
    const float* __restrict__ queries, const float* __restrict__ keysp,
    const float* __restrict__ Wq, const float* __restrict__ Wk,
    float* __restrict__ qproj, float* __restrict__ kproj)
{
    int wave = (blockIdx.x * 256 + threadIdx.x) >> 5;
    int lane = threadIdx.x & 31;

    const float *X, *W;
    float* OUT;
    int mt = wave;
    if (mt < (NB * LQ) / 16) { X = queries; W = Wq; OUT = qproj; }
    else { mt -= (NB * LQ) / 16; X = keysp; W = Wk; OUT = kproj; }

    int m0 = mt * 16;
    int lh = lane & 15;            // row (A) / col (B) within tile
    int kq = (lane >> 4) << 1;     // lanes 0-15: K=0,1 ; lanes 16-31: K=2,3

    v8f zero = {};
    v8f acc[8];
#pragma unroll
    for (int n = 0; n < 8; ++n) acc[n] = zero;

    const float* arow = X + (size_t)(m0 + lh) * QF + kq;
    for (int k0 = 0; k0 < QF; k0 += 4) {
        v2f a = *(const v2f*)(arow + k0);
#pragma unroll
        for (int n = 0; n < 8; ++n) {
            // B[k][n] = W[n][k], torch layout W:[H, QF]
            v2f bfrag = *(const v2f*)(W + (size_t)(n * 16 + lh) * QF + k0 + kq);
            acc[n] = __builtin_amdgcn_wmma_f32_16x16x4_f32(
                false, a, false, bfrag, (short)0, acc[n], false, false);
        }
    }
    // C/D layout: VGPR r -> row m0 + r + 8*(lane>=16), col = n*16 + (lane&15)
    int mrow = m0 + ((lane >> 4) << 3);
#pragma unroll
    for (int n = 0; n < 8; ++n)
#pragma unroll
        for (int r = 0; r < 8; ++r)
            OUT[(size_t)(mrow + r) * HD + n * 16 + lh] = acc[n][r];
}

// ---------------------------------------------------------------------------
// Kernel 2: raw additive-attention scores (the tanh-heavy stage).
// Block = one batch x 8 query rows. kproj[b] staged into LDS in 2 chunks of
// 256 keys, transposed with +1 padding (stride 257) for conflict-free access.
// ---------------------------------------------------------------------------
__global__ __launch_bounds__(256) void score_kernel(
    const float* __restrict__ qproj, const float* __restrict__ kproj,
    const float* __restrict__ Wv, float* __restrict__ attn)
{
    extern __shared__ float smem[];
    float* kpT = smem;                  // HD * KPAD
    float* qs  = smem + HD * KPAD;      // QROWS * HD
    float* wv  = qs + QROWS * HD;       // HD

    int tid = threadIdx.x;
    int b   = blockIdx.x >> 6;          // 64 blocks per batch (LQ/QROWS)
    int q0  = (blockIdx.x & 63) * QROWS;

    if (tid < HD) wv[tid] = Wv[tid];
    for (int idx = tid; idx < QROWS * HD; idx += 256)
        qs[idx] = qproj[(size_t)(b * LQ + q0 + (idx >> 7)) * HD + (idx & (HD - 1))];

    for (int c = 0; c < 2; ++c) {
        __syncthreads();   // protect prior-chunk reads + initial staging
        for (int idx = tid; idx < KCH * HD; idx += 256) {
            int k = idx >> 7;
            int h = idx & (HD - 1);
            kpT[h * KPAD + k] = kproj[(size_t)(b * LK + c * KCH + k) * HD + h];
        }
        __syncthreads();

        int kk = tid;  // this thread's key within the chunk
#pragma unroll
        for (int r = 0; r < QROWS; ++r) {
            const float* qr = qs + r * HD;
            float s = 0.0f;
#pragma unroll 4
            for (int h = 0; h < HD; ++h)
                s += wv[h] * tanhf(qr[h] + kpT[h * KPAD + kk]);
            attn[(size_t)(b * LQ + q0 + r) * LK + c * KCH + kk] = s;
        }
    }
}

// ---------------------------------------------------------------------------
// Kernel 3: masked softmax over the key axis, in place. One block per row.
// ---------------------------------------------------------------------------
__global__ __launch_bounds__(256) void softmax_kernel(
    float* __restrict__ attn, const int* __restrict__ valid_lens)
{
    __shared__ float red[256];
    int row = blockIdx.x;
    int b = row >> 9;                 // LQ rows per batch
    int valid = valid_lens[b];
    int tid = threadIdx.x;
    float* rp = attn + (size_t)row * LK;

    float v0 = rp[tid], v1 = rp[tid + 256];
    bool m0 = tid < valid, m1 = (tid + 256) < valid;

    float mx = fmaxf(m0 ? v0 : -3.0e38f, m1 ? v1 : -3.0e38f);
    red[tid] = mx; __syncthreads();
    for (int s = 128; s > 0; s >>= 1) {
        if (tid < s) red[tid] = fmaxf(red[tid], red[tid + s]);
        __syncthreads();
    }
    mx = red[0]; __syncthreads();

    float e0 = m0 ? __expf(v0 - mx) : 0.0f;
    float e1 = m1 ? __expf(v1 - mx) : 0.0f;
    red[tid] = e0 + e1; __syncthreads();
    for (int s = 128; s > 0; s >>= 1) {
        if (tid < s) red[tid] += red[tid + s];
        __syncthreads();
    }
    float inv = 1.0f / red[0];
    rp[tid]       = e0 * inv;
    rp[tid + 256] = e1 * inv;
}

// ---------------------------------------------------------------------------
// Kernel 4: out = attn @ values (batched, f32 WMMA 16x16x4).
// One wave computes a 16-row x 128-col tile of out.
// ---------------------------------------------------------------------------
__global__ __launch_bounds__(256) void av_kernel(
    const float* __restrict__ attn, const float* __restrict__ values,
    float* __restrict__ out)
{
    int wave = (blockIdx.x * 256 + threadIdx.x) >> 5;
    int lane = threadIdx.x & 31;

    int b   = wave >> 6;          // 64 waves per batch
    int rem = wave & 63;
    int mt  = rem >> 1;           // 32 m-tiles (512 rows / 16)
    int n0  = (rem & 1) * 128;    // two 128-wide column halves of VD=256
    int m0  = mt * 16;
    int lh  = lane & 15;
    int kq  = (lane >> 4) << 1;

    const float* A  = attn + (size_t)(b * LQ + m0 + lh) * LK + kq;
    const float* Bv = values + (size_t)b * LK * VD;

    v8f zero = {};
    v8f acc[8];
#pragma unroll
    for (int n = 0; n < 8; ++n) acc[n] = zero;

    for (int k0 = 0; k0 < LK; k0 += 4) {
        v2f a = *(const v2f*)(A + k0);
#pragma unroll
        for (int n = 0; n < 8; ++n) {
            int col = n0 + n * 16 + lh;
            v2f bfrag;
            bfrag.x = Bv[(size_t)(k0 + kq) * VD + col];
            bfrag.y = Bv[(size_t)(k0 + kq + 1) * VD + col];
            acc[n] = __builtin_amdgcn_wmma_f32_16x16x4_f32(
                false, a, false, bfrag, (short)0, acc[n], false, false);
        }
    }
    int mrow = m0 + ((lane >> 4) << 3);
#pragma unroll
    for (int n = 0; n < 8; ++n)
#pragma unroll
        for (int r = 0; r < 8; ++r)
            out[(size_t)(b * LQ + mrow + r) * VD + n0 + n * 16 + lh] = acc[n][r];
}

// ---------------------------------------------------------------------------
extern "C" void kernel_launch(void* const* d_in, const int* in_sizes, int n_in,
                              void* d_out, int out_size, void* d_ws, size_t ws_size,
                              hipStream_t stream) {
    const float* queries = (const float*)d_in[0];
    const float* keysp   = (const float*)d_in[1];
    const float* values  = (const float*)d_in[2];
    const int*   vlens   = (const int*)d_in[3];
    const float* Wq      = (const float*)d_in[4];
    const float* Wk      = (const float*)d_in[5];
    const float* Wv      = (const float*)d_in[6];

    float* out   = (float*)d_out;                    // [B, LQ, VD]
    float* attn  = out + (size_t)NB * LQ * VD;       // [B, LQ, LK]
    float* qproj = (float*)d_ws;                     // [B*LQ, H]
    float* kproj = qproj + (size_t)NB * LQ * HD;     // [B*LK, H]

    // Stage 1: projections (256 waves -> 32 blocks of 8 waves)
    proj_kernel<<<32, 256, 0, stream>>>(queries, keysp, Wq, Wk, qproj, kproj);

    // Stage 2: scores (dynamic LDS ~133 KB, above the 64 KB default)
    size_t smem_bytes = (size_t)(HD * KPAD + QROWS * HD + HD) * sizeof(float);
    hipFuncSetAttribute((const void*)score_kernel,
                        hipFuncAttributeMaxDynamicSharedMemorySize,
                        (int)smem_bytes);
    score_kernel<<<NB * (LQ / QROWS), 256, smem_bytes, stream>>>(qproj, kproj, Wv, attn);

    // Stage 3: masked softmax, one block per (b, q) row
    softmax_kernel<<<NB * LQ, 256, 0, stream>>>(attn, vlens);

    // Stage 4: out = attn @ values (256 waves -> 32 blocks)
    av_kernel<<<32, 256, 0, stream>>>(attn, values, out);
}